// Exchange_11055245820589
// MI455X (gfx1250) — compile-verified
//
#include <hip/hip_runtime.h>
#include <cmath>

typedef __attribute__((ext_vector_type(2))) float v2f;
typedef __attribute__((ext_vector_type(8))) float v8f;

#define NN      16384
#define L0DIM   128
#define HDIM    64
#define R2      64.0f   // RADIUS^2

// ---------------------------------------------------------------------------
// Kernel 1: out = silu(emb[z] @ w1 + b1) @ w2 + b2     (one wave = 16 rows)
// Uses V_WMMA_F32_16X16X4_F32 tiles: A 16x4 f32, B 4x16 f32, C/D 16x16 f32.
// ---------------------------------------------------------------------------
__global__ __launch_bounds__(256) void mlp_wmma_kernel(
    const int*   __restrict__ z,
    const float* __restrict__ emb,
    const float* __restrict__ w1,   // [128][64] row-major
    const float* __restrict__ b1,   // [64]
    const float* __restrict__ w2,   // [64]
    const float* __restrict__ b2,   // [1]
    float*       __restrict__ out)  // [N]
{
    const int lane = threadIdx.x & 31;
    const int wave = (blockIdx.x * blockDim.x + threadIdx.x) >> 5;
    const int m0   = wave * 16;

    const int half = lane >> 4;   // 0: K pair {0,1}, 1: K pair {2,3}
    const int l16  = lane & 15;

    // A-side: this lane supplies row m = m0 + l16 of the gathered embedding
    const int   zrow = z[m0 + l16];
    const float* arow = emb + (long)zrow * L0DIM;

    v8f acc[4];
#pragma unroll
    for (int nt = 0; nt < 4; ++nt) acc[nt] = v8f{};

    for (int kk = 0; kk < L0DIM; kk += 4) {
        const int k0 = kk + 2 * half;
        v2f a;
        a.x = arow[k0 + 0];
        a.y = arow[k0 + 1];

        const float* brow0 = w1 + (long)(k0 + 0) * HDIM;
        const float* brow1 = w1 + (long)(k0 + 1) * HDIM;
#pragma unroll
        for (int nt = 0; nt < 4; ++nt) {
            v2f b;
            b.x = brow0[nt * 16 + l16];
            b.y = brow1[nt * 16 + l16];
            acc[nt] = __builtin_amdgcn_wmma_f32_16x16x4_f32(
                false, a, false, b, (short)0, acc[nt], false, false);
        }
    }

    // Epilogue: h = silu(acc + b1); p = h . w2 ; reduce over 16 cols twice
    float b1c[4], w2c[4];
#pragma unroll
    for (int nt = 0; nt < 4; ++nt) {
        b1c[nt] = b1[nt * 16 + l16];
        w2c[nt] = w2[nt * 16 + l16];
    }
    const float bias2 = b2[0];

#pragma unroll
    for (int r = 0; r < 8; ++r) {
        float p = 0.0f;
#pragma unroll
        for (int nt = 0; nt < 4; ++nt) {
            float x = acc[nt][r] + b1c[nt];
            // silu(x) = x * sigmoid(x), fast rcp (v_rcp_f32) instead of
            // the IEEE div_scale/fixup sequence
            float s = x * __builtin_amdgcn_rcpf(1.0f + __expf(-x));
            p += s * w2c[nt];
        }
        // reduce across the 16 lanes of each half-wave (cols 0..63 total)
        p += __shfl_xor(p, 1, 32);
        p += __shfl_xor(p, 2, 32);
        p += __shfl_xor(p, 4, 32);
        p += __shfl_xor(p, 8, 32);
        if (l16 == 0) out[m0 + r + 8 * half] = p + bias2;
    }
}

// ---------------------------------------------------------------------------
// Kernel 2: adj[i][j] = (sq_i + sq_j - 2*dot(pos_i,pos_j) < R2)
//                        && batch_i==batch_j && i!=j     (float 0/1)
// One wave owns a 16-wide j-tile; loops over 16 i-tiles. pos@pos^T via
// V_WMMA_F32_16X16X4_F32 with K=3 zero-padded to 4. Store-bandwidth bound:
// 1 GB of pure streaming writes -> non-temporal stores so the stream does
// not evict the hot pos/batch working set from WGP$/L2.
// ---------------------------------------------------------------------------
__global__ __launch_bounds__(256) void adj_wmma_kernel(
    const float* __restrict__ pos,    // [N][3]
    const int*   __restrict__ batch,  // [N]
    float*       __restrict__ adj)    // [N][N]
{
    const int lane = threadIdx.x & 31;
    const int half = lane >> 4;
    const int l16  = lane & 15;

    const int waveInBlock = threadIdx.x >> 5;
    const int j0 = (blockIdx.x * 8 + waveInBlock) * 16;  // j-tile
    const int iBase = blockIdx.y * 256;                  // 16 i-tiles of 16

    // B-side (j columns), loaded once: B[k][n] = pos[j0+n][k], k=3 padded 0
    const int   jn  = j0 + l16;
    const float pjx = pos[3 * jn + 0];
    const float pjy = pos[3 * jn + 1];
    const float pjz = pos[3 * jn + 2];
    const float sqj = pjx * pjx + pjy * pjy + pjz * pjz;
    const int   bj  = batch[jn];

    v2f b;
    if (half == 0) { b.x = pjx; b.y = pjy; }
    else           { b.x = pjz; b.y = 0.0f; }

    for (int t = 0; t < 16; ++t) {
        const int i0 = iBase + t * 16;

        // A-side: lane supplies row m = l16 of the i-tile
        const int   im  = i0 + l16;
        const float pix = pos[3 * im + 0];
        const float piy = pos[3 * im + 1];
        const float piz = pos[3 * im + 2];
        const float sqi = pix * pix + piy * piy + piz * piz;
        const int   bi  = batch[im];

        v2f a;
        if (half == 0) { a.x = pix; a.y = piy; }
        else           { a.x = piz; a.y = 0.0f; }

        v8f dot = {};
        dot = __builtin_amdgcn_wmma_f32_16x16x4_f32(
            false, a, false, b, (short)0, dot, false, false);

        // D element (lane, vgpr r) -> row M = r + 8*half, col N = l16
#pragma unroll
        for (int r = 0; r < 8; ++r) {
            const int srcLane = r + 8 * half;         // rows live in lanes 0..15
            const float sqi_r = __shfl(sqi, srcLane, 32);
            const int   bi_r  = __shfl(bi,  srcLane, 32);
            const int   iRow  = i0 + srcLane;

            const float d2 = sqi_r + sqj - 2.0f * dot[r];
            const bool ok = (d2 < R2) && (bi_r == bj) && (iRow != jn);
            __builtin_nontemporal_store(ok ? 1.0f : 0.0f,
                                        adj + (long)iRow * NN + jn);
        }
    }
}

extern "C" void kernel_launch(void* const* d_in, const int* in_sizes, int n_in,
                              void* d_out, int out_size, void* d_ws, size_t ws_size,
                              hipStream_t stream) {
    const int*   z     = (const int*)  d_in[0];
    const int*   batch = (const int*)  d_in[1];   // JAX default demotes int64->int32
    const float* pos   = (const float*)d_in[2];
    const float* emb   = (const float*)d_in[3];
    const float* w1    = (const float*)d_in[4];
    const float* b1    = (const float*)d_in[5];
    const float* w2    = (const float*)d_in[6];
    const float* b2    = (const float*)d_in[7];

    float* out = (float*)d_out;          // [N] scores
    float* adj = out + NN;               // [N][N] mask, row-major

    // MLP: 1024 waves (16 rows each), 8 waves per block -> 128 blocks
    mlp_wmma_kernel<<<dim3(128), dim3(256), 0, stream>>>(
        z, emb, w1, b1, w2, b2, out);

    // Adjacency: grid.x covers 1024 j-tiles (8 waves/block), grid.y = 64
    // chunks of 256 rows each.
    adj_wmma_kernel<<<dim3(128, 64), dim3(256), 0, stream>>>(pos, batch, adj);
}